// AEVComputer_87883620811433
// MI455X (gfx1250) — compile-verified
//
#include <hip/hip_runtime.h>
#include <math.h>

// ---- problem constants (match reference) ----
#define CN 2
#define AN 96
#define SN 4
#define NNB 95                 // neighbors per atom
#define NPAIRS 4465            // C(95,2)
#define TILE 64                // pairs per staging tile
#define NTILES 70              // 70*64 = 4480 >= 4465
#define RCR_C 5.2f
#define RCA_C 3.5f
#define NFEAT 384              // 64 radial + 320 angular
#define WPB 8                  // waves per block (256 threads)
#define LOG2E 1.4426950408889634f

typedef __attribute__((ext_vector_type(2))) float v2f;
typedef __attribute__((ext_vector_type(8))) float v8f;

__device__ __forceinline__ int species_pair_index(int s1, int s2) {
    int lo = min(s1, s2), hi = max(s1, s2);
    return lo * SN - (lo * (lo + 1)) / 2 + hi;   // (0,0)=0 .. (3,3)=9
}

__global__ __launch_bounds__(WPB * 32)
void aev_kernel(const int* __restrict__ species,
                const float* __restrict__ coords,
                const float* __restrict__ EtaR,
                const float* __restrict__ ShfR,
                const float* __restrict__ EtaA,
                const float* __restrict__ Zeta,
                const float* __restrict__ ShfA,
                const float* __restrict__ ShfZ,
                float* __restrict__ out) {
    // ---- LDS ----
    __shared__ unsigned short s_pairs[NPAIRS];        // j1 | (j2<<8)  (block-shared)
    __shared__ float4 s_nb4[WPB][NNB + 1];            // {vx, vy, vz, d}
    __shared__ float2 s_nbm[WPB][NNB + 1];            // {fcA, fcR}
    __shared__ int    s_sp [WPB][NNB + 1];            // neighbor species
    __shared__ float4 s_ps [WPB][2][TILE];            // pair scalars {cos,sin,dsum,2*fcj}
    __shared__ int    s_pi [WPB][2][TILE];            // pair species-pair index

    const int tid  = threadIdx.x;
    const int lane = tid & 31;
    const int wv   = tid >> 5;                        // wave in block
    const int wid  = blockIdx.x * WPB + wv;           // global wave == atom index
    const int ci   = wid / AN;                        // conformer
    const int ai   = wid % AN;                        // atom

    // ---- build universal pair table once per block ----
    if (tid < NNB) {
        const int j1 = tid;
        int off = 94 * j1 - (j1 * (j1 - 1)) / 2;      // row start in triu(95,1)
        for (int j2 = j1 + 1; j2 < NNB; ++j2)
            s_pairs[off++] = (unsigned short)(j1 | (j2 << 8));
    }

    // ---- per-wave: stage neighbor data for this atom into LDS ----
    const float xi = coords[(ci * AN + ai) * 3 + 0];
    const float yi = coords[(ci * AN + ai) * 3 + 1];
    const float zi = coords[(ci * AN + ai) * 3 + 2];
    for (int idx = lane; idx < NNB; idx += 32) {
        const int j = idx + (idx >= ai ? 1 : 0);      // nb[i] skips i
        const float dx = xi - coords[(ci * AN + j) * 3 + 0];
        const float dy = yi - coords[(ci * AN + j) * 3 + 1];
        const float dz = zi - coords[(ci * AN + j) * 3 + 2];
        const float d  = __builtin_amdgcn_sqrtf(dx * dx + dy * dy + dz * dz);
        s_nb4[wv][idx] = make_float4(dx, dy, dz, d);
        const float fa = (d <= RCA_C) ? (0.5f * cosf((float)M_PI * d / RCA_C) + 0.5f) : 0.0f;
        const float fr = (d <= RCR_C) ? (0.5f * cosf((float)M_PI * d / RCR_C) + 0.5f) : 0.0f;
        s_nbm[wv][idx] = make_float2(fa, fr);
        s_sp [wv][idx] = species[ci * AN + j];
    }
    __syncthreads();

    const float etaR = EtaR[0];
    const float etaA = EtaA[0];
    const float zeta = Zeta[0];
    const float nErL2 = -etaR * LOG2E;
    const float nEaL2 = -etaA * LOG2E;

    // =====================  RADIAL (VALU, 64 features)  =====================
    // lane owns features f=lane (s=lane>>4, r=lane&15) and f=lane+32 (s+2, r)
    {
        const int s0 = lane >> 4, r0 = lane & 15, s1 = s0 + 2;
        const float shr = ShfR[r0];
        float acc0 = 0.0f, acc1 = 0.0f;
        for (int j = 0; j < NNB; ++j) {
            const float d  = s_nb4[wv][j].w;
            const float fc = s_nbm[wv][j].y;
            const int   sj = s_sp [wv][j];
            const float dd = d - shr;
            const float g  = 0.25f * __builtin_amdgcn_exp2f(nErL2 * dd * dd) * fc;
            acc0 += (sj == s0) ? g : 0.0f;
            acc1 += (sj == s1) ? g : 0.0f;
        }
        float* outR = out + (size_t)(ci * AN + ai) * NFEAT;
        outR[lane]      = acc0;
        outR[lane + 32] = acc1;
    }

    // ================  ANGULAR (WMMA f32 16x16x4, 320 features)  ================
    // Two-phase tiles: phase1 computes per-pair scalars ONCE (no 16x redundancy),
    // phase2 runs K=4 WMMA chunks: D(16x16) += maskT(16x4) * aterms(4x16).
    //   A rows = species-pair bins (10 of 16); B cols = features (two 16-wide tiles)
    //   f32 WMMA layouts: lanes 0-15 hold K=0,1; lanes 16-31 hold K=2,3.
    const int n   = lane & 15;           // feature column (and A-matrix row)
    const int ia  = n >> 3;              // ShfA index for tile0 (tile1: ia+2)
    const int iz  = n & 7;               // ShfZ index (same for both tiles)
    const float shfz = ShfZ[iz];
    const float cz = cosf(shfz), sz = sinf(shfz);
    const float shfa0 = ShfA[ia];
    const float shfa1 = ShfA[ia + 2];
    const int koff = (lane < 16) ? 0 : 2;

    v8f c0 = {};   // features 0..15
    v8f c1 = {};   // features 16..31
    int buf = 0;
    for (int t = 0; t < NTILES; ++t, buf ^= 1) {
        // ---- phase 1: per-pair feature-independent scalars (2 iters of 32 lanes) ----
#pragma unroll
        for (int it = 0; it < TILE / 32; ++it) {
            const int q = it * 32 + lane;
            const int p = t * TILE + q;
            float ca = 0.0f, sa = 0.0f, dsum = 0.0f, w = 0.0f;
            int spidx = -1;
            if (p < NPAIRS) {
                const unsigned int pk = s_pairs[p];
                const int j1 = pk & 0xFF;
                const int j2 = pk >> 8;
                const float4 n1 = s_nb4[wv][j1];
                const float4 n2 = s_nb4[wv][j2];
                const float dot = n1.x * n2.x + n1.y * n2.y + n1.z * n2.z;
                const float d1 = n1.w, d2 = n2.w;
                ca = 0.95f * dot *
                     __builtin_amdgcn_rcpf(fmaxf(d1, 1e-8f) * fmaxf(d2, 1e-8f));
                sa = __builtin_amdgcn_sqrtf(fmaxf(1.0f - ca * ca, 0.0f));
                dsum = 0.5f * (d1 + d2);
                w = 2.0f * s_nbm[wv][j1].x * s_nbm[wv][j2].x;   // 2*fcj
                spidx = species_pair_index(s_sp[wv][j1], s_sp[wv][j2]);
            }
            s_ps[wv][buf][q] = make_float4(ca, sa, dsum, w);
            s_pi[wv][buf][q] = spidx;
        }
        __syncthreads();   // uniform: all waves run identical tile counts

        // ---- phase 2: 16 WMMA chunks (K=4 pairs each) ----
        for (int ch = 0; ch < TILE / 4; ++ch) {
            float Av[2], B0v[2], B1v[2];
#pragma unroll
            for (int u = 0; u < 2; ++u) {
                const int q = ch * 4 + koff + u;
                const float4 ps = s_ps[wv][buf][q];
                const int spidx = s_pi[wv][buf][q];
                // cos(acos(ca) - shfz) = ca*cz + sa*sz
                const float cdiff = ps.x * cz + ps.y * sz;
                const float tbase = 0.5f + 0.5f * cdiff;        // in [0,1]
                const float f1 = __builtin_amdgcn_exp2f(
                                     zeta * __builtin_amdgcn_logf(tbase));
                const float wf1 = ps.w * f1;                    // 2*fcj*f1
                const float e0 = ps.z - shfa0;
                const float e1 = ps.z - shfa1;
                B0v[u] = wf1 * __builtin_amdgcn_exp2f(nEaL2 * e0 * e0);
                B1v[u] = wf1 * __builtin_amdgcn_exp2f(nEaL2 * e1 * e1);
                Av[u]  = (spidx == n) ? 1.0f : 0.0f;
            }
            v2f av, b0, b1;
            av.x = Av[0];  av.y = Av[1];
            b0.x = B0v[0]; b0.y = B0v[1];
            b1.x = B1v[0]; b1.y = B1v[1];
            // (neg_a, A, neg_b, B, c_mod, C, reuse_a, reuse_b)
            c0 = __builtin_amdgcn_wmma_f32_16x16x4_f32(false, av, false, b0,
                                                       (short)0, c0, false, false);
            c1 = __builtin_amdgcn_wmma_f32_16x16x4_f32(false, av, false, b1,
                                                       (short)0, c1, false, false);
        }
    }

    // D layout: VGPR r -> M = r (lanes 0-15) / r+8 (lanes 16-31), N = lane&15
    {
        float* outA = out + (size_t)(ci * AN + ai) * NFEAT + 64;
        const int mbase = (lane < 16) ? 0 : 8;
        const int feat  = lane & 15;
#pragma unroll
        for (int r = 0; r < 8; ++r) {
            const int sp = mbase + r;
            if (sp < 10) {
                outA[sp * 32 + feat]      = c0[r];
                outA[sp * 32 + 16 + feat] = c1[r];
            }
        }
    }
}

extern "C" void kernel_launch(void* const* d_in, const int* in_sizes, int n_in,
                              void* d_out, int out_size, void* d_ws, size_t ws_size,
                              hipStream_t stream) {
    const int*   species = (const int*)  d_in[0];   // (C, A)
    const float* coords  = (const float*)d_in[1];   // (C, A, 3)
    const float* EtaR    = (const float*)d_in[2];   // (1,)
    const float* ShfR    = (const float*)d_in[3];   // (16,)
    const float* EtaA    = (const float*)d_in[4];   // (1,)
    const float* Zeta    = (const float*)d_in[5];   // (1,)
    const float* ShfA    = (const float*)d_in[6];   // (4,)
    const float* ShfZ    = (const float*)d_in[7];   // (8,)
    float* out = (float*)d_out;                      // (C, A, 384)

    const int total_waves = CN * AN;                 // 192, one wave32 per atom
    const int blocks = total_waves / WPB;            // 24 blocks of 256 threads
    aev_kernel<<<blocks, WPB * 32, 0, stream>>>(species, coords, EtaR, ShfR,
                                                EtaA, Zeta, ShfA, ShfZ, out);
}